// CoBertCRF_68436008894605
// MI455X (gfx1250) — compile-verified
//
#include <hip/hip_runtime.h>
#include <hip/hip_bf16.h>
#include <math.h>

// Problem constants (from reference).
#define BB 128
#define SS 512
#define HH 1024
#define TT 13
#define NCHUNK (HH / 32)

typedef __attribute__((ext_vector_type(2))) float v2f;
typedef __attribute__((ext_vector_type(8))) float v8f;
typedef __attribute__((ext_vector_type(4))) unsigned int v4u;
typedef __attribute__((ext_vector_type(8))) int v8i;
typedef __attribute__((ext_vector_type(4))) int v4i;

#if __has_builtin(__builtin_amdgcn_tensor_load_to_lds) && \
    __has_builtin(__builtin_amdgcn_s_wait_tensorcnt)
#define HAVE_TDM 1
#else
#define HAVE_TDM 0
#endif

#if HAVE_TDM
// ---------------------------------------------------------------------------
// Issue one TDM transfer: 128 rows x 32 f32 tile of `hidden` -> LDS, with
// hardware row padding (+4 DWORDs per 32 DWORDs) => LDS row stride 36 floats.
// Descriptor per CDNA5 ISA ch.8 (D# groups). Must be executed by one wave.
// This toolchain exposes the 6-arg builtin: (g0 v4u, g1 v8i, g2 v4i, g3 v4i,
// extra v8i, i32 cpol); groups 2/3/extra are zero for a 2D tile.
// ---------------------------------------------------------------------------
__device__ __forceinline__ void tdm_load_tile(const float* gsrc,
                                              unsigned lds_byte_off) {
  const unsigned long long ga = (unsigned long long)(uintptr_t)gsrc;
  v4u g0;
  g0[0] = 1u;                                   // count=1 (valid), user mode
  g0[1] = lds_byte_off;                         // lds_addr (bytes)
  g0[2] = (unsigned)(ga & 0xFFFFFFFFu);         // global_addr[31:0]
  g0[3] = (unsigned)((ga >> 32) & 0x01FFFFFFu)  // global_addr[56:32]
        | (2u << 30);                           // type=2 ("image")
  v8i g1;
  g1[0] = (int)((2u << 16)      // data_size = 4 bytes
              | (1u << 20)      // pad_enable
              | (4u << 22)      // pad_interval: code 4 = every 32 DWORDs
              | (3u << 25));    // pad_amount:  code 3 = 4 DWORDs
  g1[1] = (int)(1024u << 16);   // tensor_dim0 = 1024 (bits[79:48], low half)
  g1[2] = 0;                    // tensor_dim0 hi16 = 0; tensor_dim1 lo16 = 0
  g1[3] = (int)(1u              // tensor_dim1 hi16 -> dim1 = 65536 rows
              | (32u << 16));   // tile_dim0 = 32
  g1[4] = 128;                  // tile_dim1 = 128 rows; tile_dim2 = 0
  g1[5] = 1024;                 // tensor_dim0_stride = 1024 (lo32)
  g1[6] = 0;                    // stride hi16; tensor_dim1_stride lo16
  g1[7] = 0;
  const v4i z4 = {0, 0, 0, 0};              // groups 2/3 unused (2D tile)
  const v8i z8 = {0, 0, 0, 0, 0, 0, 0, 0};  // extra group (6-arg variant)
  __builtin_amdgcn_tensor_load_to_lds(g0, g1, z4, z4, z8, 0);
}
#endif

// ---------------------------------------------------------------------------
// Kernel 1: token_scores[b,s,t] = sum_h hidden[b,s,h]*W[t,h] + bias[t]
// GEMM (65536 x 13 x 1024) via V_WMMA_F32_16X16X4_F32.
//   Block = 256 threads = 8 waves; block covers 128 tokens (8 M-tiles).
//   K in chunks of 32, double-buffered:
//     A tile : TDM (tensor_load_to_lds) -> LDS, padded stride 36 floats
//     B tile : W chunk (32x16, zero-padded past T=13), regs -> LDS
// ---------------------------------------------------------------------------
__global__ __launch_bounds__(256)
void gemm_scores_kernel(const float* __restrict__ hidden,
                        const float* __restrict__ W,
                        const float* __restrict__ bias,
                        float* __restrict__ out_scores) {
  __shared__ float sA[2][128][36];   // 36*4 = 144B row => 16B-aligned, no bank
  __shared__ float sB[2][32][16];    // conflicts on wmma-pattern reads

  const int tid  = threadIdx.x;
  const int lane = tid & 31;
  const int wave = tid >> 5;          // 0..7
  const int m    = lane & 15;
  const int half = lane >> 4;         // K-split half per ISA A/B layouts

  const int tok0 = blockIdx.x * 128;  // 512 blocks * 128 tokens

  // W element indices owned by this thread (2 per chunk)
  const int e0 = tid,        k0w = e0 >> 4, n0w = e0 & 15;
  const int e1 = tid + 256,  k1w = e1 >> 4, n1w = e1 & 15;

  v8f acc = {};

  // ---------------- prologue: stage chunk 0 --------------------------------
  {
    const float w0 = (n0w < TT) ? W[(size_t)n0w * HH + k0w] : 0.0f;
    const float w1 = (n1w < TT) ? W[(size_t)n1w * HH + k1w] : 0.0f;
    sB[0][k0w][n0w] = w0;
    sB[0][k1w][n1w] = w1;
#if HAVE_TDM
    if (wave == 0) {
      tdm_load_tile(hidden + (size_t)tok0 * HH,
                    (unsigned)(uintptr_t)&sA[0][0][0]);
      __builtin_amdgcn_s_wait_tensorcnt(0);
    }
#else
    #pragma unroll
    for (int i = 0; i < 4; ++i) {
      int f  = tid + i * 256;
      int t  = f >> 3;
      int kq = (f & 7) << 2;
      const float4 v = *reinterpret_cast<const float4*>(
          hidden + (size_t)(tok0 + t) * HH + kq);
      *reinterpret_cast<float4*>(&sA[0][t][kq]) = v;
    }
#endif
    __syncthreads();
  }

  // ---------------- main pipelined loop ------------------------------------
  for (int c = 0; c < NCHUNK; ++c) {
    const int cur = c & 1;
    const int nxt = cur ^ 1;
    const bool more = (c + 1) < NCHUNK;

    float w0 = 0.f, w1 = 0.f;
    if (more) {
      const int kk = (c + 1) * 32;
      // next W chunk into registers (latency hidden under WMMAs)
      w0 = (n0w < TT) ? W[(size_t)n0w * HH + kk + k0w] : 0.0f;
      w1 = (n1w < TT) ? W[(size_t)n1w * HH + kk + k1w] : 0.0f;
#if HAVE_TDM
      // DMA next A tile; buffer `nxt` was last read at c-1 (barrier-fenced)
      if (wave == 0) {
        tdm_load_tile(hidden + (size_t)tok0 * HH + kk,
                      (unsigned)(uintptr_t)&sA[nxt][0][0]);
      }
#else
      #pragma unroll
      for (int i = 0; i < 4; ++i) {
        int f  = tid + i * 256;
        int t  = f >> 3;
        int kq = (f & 7) << 2;
        const float4 v = *reinterpret_cast<const float4*>(
            hidden + (size_t)(tok0 + t) * HH + kk + kq);
        *reinterpret_cast<float4*>(&sA[nxt][t][kq]) = v;
      }
#endif
    }

    // ---- 8 x V_WMMA_F32_16X16X4_F32 on current chunk ----
    const int arow = wave * 16 + m;
    #pragma unroll
    for (int kq = 0; kq < 32; kq += 4) {
      const int ka = kq + half * 2;
      v2f a, bmat;
      a.x = sA[cur][arow][ka];       // A 16x4: lane(m,half) holds K=ka,ka+1
      a.y = sA[cur][arow][ka + 1];
      bmat.x = sB[cur][ka][m];       // B 4x16: row-striped across lanes
      bmat.y = sB[cur][ka + 1][m];
      acc = __builtin_amdgcn_wmma_f32_16x16x4_f32(
          false, a, false, bmat, (short)0, acc, false, false);
    }

    if (more) {
      sB[nxt][k0w][n0w] = w0;        // other waves only touch `cur` buffers
      sB[nxt][k1w][n1w] = w1;
#if HAVE_TDM
      if (wave == 0) __builtin_amdgcn_s_wait_tensorcnt(0);
#endif
    }
    __syncthreads();                 // releases `nxt` buffers for iter c+1
  }

  // ---- epilogue: C/D layout (VGPR r: M = r + half*8, N = m) -> global ----
  if (m < TT) {
    const float bv = bias[m];
    #pragma unroll
    for (int r = 0; r < 8; ++r) {
      const size_t token = (size_t)tok0 + wave * 16 + r + half * 8;
      out_scores[token * TT + m] = acc[r] + bv;
    }
  }
}

// ---------------------------------------------------------------------------
// Kernel 2: CRF forward (logsumexp) + Viterbi per batch element.
//   One block (1 wave) per batch element; lanes 0..12 = tags.
// ---------------------------------------------------------------------------
__global__ __launch_bounds__(32)
void crf_kernel(const float* __restrict__ scores,   // [B,S,T]
                const float* __restrict__ start_trans,
                const float* __restrict__ end_trans,
                const float* __restrict__ trans,    // [T,T]
                const int*   __restrict__ labels,   // [B,S]
                const int*   __restrict__ mask,     // [B,S]
                float* __restrict__ out_tags,       // [B,S] (float)
                float* __restrict__ ws_llh,         // [B]
                int*   __restrict__ ws_hist) {      // [B, S-1, T]
  __shared__ float alpha[16];
  __shared__ float vit[16];

  const int b = blockIdx.x;
  const int j = threadIdx.x;
  const bool act = (j < TT);
  const float* Sc = scores + (size_t)b * SS * TT;
  int* hist = ws_hist + (size_t)b * (SS - 1) * TT;

  float trc[TT];
  if (act) {
    #pragma unroll
    for (int i = 0; i < TT; ++i) trc[i] = trans[i * TT + j];
    const float init = start_trans[j] + Sc[j];
    alpha[j] = init;
    vit[j]   = init;
  }
  __syncthreads();

  for (int s = 1; s < SS; ++s) {
    float aold = 0.f, vold = 0.f, nxt = 0.f, vnxt = 0.f;
    int barg = 0;
    const int msk = mask[(size_t)b * SS + s];
    if (act) {
      const float em = Sc[(size_t)s * TT + j];
      aold = alpha[j];
      vold = vit[j];
      float mx = -INFINITY;
      float cv[TT];
      #pragma unroll
      for (int i = 0; i < TT; ++i) {
        cv[i] = alpha[i] + trc[i];
        mx = fmaxf(mx, cv[i]);
      }
      float sum = 0.f;
      #pragma unroll
      for (int i = 0; i < TT; ++i) sum += __expf(cv[i] - mx);
      nxt = mx + __logf(sum) + em;
      float bmax = -INFINITY;
      #pragma unroll
      for (int i = 0; i < TT; ++i) {
        const float c = vit[i] + trc[i];
        if (c > bmax) { bmax = c; barg = i; }   // first max (jnp.argmax)
      }
      vnxt = bmax + em;
    }
    __syncthreads();
    if (act) {
      alpha[j] = msk ? nxt : aold;
      vit[j]   = msk ? vnxt : vold;
      hist[(size_t)(s - 1) * TT + j] = msk ? barg : j;
    }
    __syncthreads();
  }

  if (j == 0) {
    // denominator: logsumexp(alpha + end_trans)
    float mx = -INFINITY;
    for (int i = 0; i < TT; ++i) mx = fmaxf(mx, alpha[i] + end_trans[i]);
    float sum = 0.f;
    for (int i = 0; i < TT; ++i) sum += __expf(alpha[i] + end_trans[i] - mx);
    const float denom = mx + __logf(sum);

    // numerator (raw label chain, mask-weighted)
    const int* lb = labels + (size_t)b * SS;
    const int* mk = mask   + (size_t)b * SS;
    int prev = lb[0];
    float num = start_trans[prev] + Sc[prev];
    int seq_end = 0;
    for (int s = 1; s < SS; ++s) {
      const int l = lb[s];
      const float mf = mk[s] ? 1.0f : 0.0f;
      num += (trans[prev * TT + l] + Sc[(size_t)s * TT + l]) * mf;
      prev = l;
      seq_end += mk[s] ? 1 : 0;
    }
    num += end_trans[lb[seq_end]];
    ws_llh[b] = num - denom;

    // Viterbi backtrack
    float bmax = -INFINITY;
    int tag = 0;
    for (int i = 0; i < TT; ++i) {
      const float c = vit[i] + end_trans[i];
      if (c > bmax) { bmax = c; tag = i; }
    }
    float* ot = out_tags + (size_t)b * SS;
    ot[SS - 1] = mk[SS - 1] ? (float)tag : 0.0f;
    for (int s = SS - 2; s >= 0; --s) {
      tag = hist[(size_t)s * TT + tag];
      ot[s] = mk[s] ? (float)tag : 0.0f;
    }
  }
}

// ---------------------------------------------------------------------------
// Kernel 3: deterministic final reduction: loss = -sum(llh)/B
// ---------------------------------------------------------------------------
__global__ void finalize_kernel(const float* __restrict__ ws_llh,
                                float* __restrict__ d_out) {
  if (threadIdx.x == 0 && blockIdx.x == 0) {
    float s = 0.f;
    for (int b = 0; b < BB; ++b) s += ws_llh[b];
    d_out[0] = -s / (float)BB;
  }
}

// ---------------------------------------------------------------------------
extern "C" void kernel_launch(void* const* d_in, const int* in_sizes, int n_in,
                              void* d_out, int out_size, void* d_ws, size_t ws_size,
                              hipStream_t stream) {
  const float* hidden      = (const float*)d_in[0];   // [B,S,H]
  const float* W           = (const float*)d_in[1];   // [T,H]
  const float* bias        = (const float*)d_in[2];   // [T]
  const float* start_trans = (const float*)d_in[3];   // [T]
  const float* end_trans   = (const float*)d_in[4];   // [T]
  const float* trans       = (const float*)d_in[5];   // [T,T]
  const int*   labels      = (const int*)d_in[6];     // [B,S]
  const int*   mask        = (const int*)d_in[7];     // [B,S]

  float* out        = (float*)d_out;
  float* out_tags   = out + 1;                        // [B,S] as float
  float* out_scores = out + 1 + (size_t)BB * SS;      // [B,S,T]

  float* ws_llh  = (float*)d_ws;                      // [B]
  int*   ws_hist = (int*)((char*)d_ws + 1024);        // [B,S-1,T]

  gemm_scores_kernel<<<(BB * SS) / 128, 256, 0, stream>>>(hidden, W, bias,
                                                          out_scores);
  crf_kernel<<<BB, 32, 0, stream>>>(out_scores, start_trans, end_trans, trans,
                                    labels, mask, out_tags, ws_llh, ws_hist);
  finalize_kernel<<<1, 1, 0, stream>>>(ws_llh, out);
}